// MF_1331439862348
// MI455X (gfx1250) — compile-verified
//
#include <hip/hip_runtime.h>

typedef float v2f __attribute__((ext_vector_type(2)));
typedef float v8f __attribute__((ext_vector_type(8)));

#define NF 64  // factors per embedding row

// One wave computes 16 batch elements: D(16x16) = UE(16x64) x UI^T(64x16) via
// 16 steps of v_wmma_f32_16x16x4_f32; diag(D) = the 16 dot products.
__global__ __launch_bounds__(256) void mf_wmma_dot_kernel(
    const int*   __restrict__ users,
    const int*   __restrict__ items,
    const float* __restrict__ user_emb,
    const float* __restrict__ item_emb,
    const float* __restrict__ user_bias,
    const float* __restrict__ item_bias,
    const float* __restrict__ bias,
    float*       __restrict__ out,
    int batch)
{
    const int lane = threadIdx.x & 31;
    const int wave = threadIdx.x >> 5;
    const int base = blockIdx.x * 128 + wave * 16;   // 16 rows per wave
    if (base >= batch) return;                        // uniform per wave

    // This lane feeds row (M for A, N for B) = lane%16; lanes>=16 carry K+2..K+3.
    const int row   = lane & 15;
    const int khalf = (lane >> 4) << 1;              // 0 or 2

    int b = base + row;
    if (b >= batch) b = batch - 1;                   // safety clamp (BATCH%16==0 anyway)
    const int uidx = users[b];
    const int iidx = items[b];
    const float* __restrict__ urow = user_emb + (size_t)uidx * NF;
    const float* __restrict__ irow = item_emb + (size_t)iidx * NF;

    v8f c = {};                                      // 16x16 f32 accumulator
#pragma unroll
    for (int k = 0; k < NF; k += 4) {
        // A: lane holds (M=row, K=k+khalf, k+khalf+1)
        v2f a  = *(const v2f*)(urow + k + khalf);
        // B: lane holds (N=row, K=k+khalf, k+khalf+1); B[k][n] = irow_n[k]
        v2f bb = *(const v2f*)(irow + k + khalf);
        // 8 args: (neg_a, A, neg_b, B, c_mod, C, reuse_a, reuse_b)
        c = __builtin_amdgcn_wmma_f32_16x16x4_f32(false, a, false, bb,
                                                  (short)0, c, false, false);
    }

    // Diagonal extraction from C/D layout:
    //   lanes 0-15 : c[v] = D[M=v    ][N=lane]
    //   lanes 16-31: c[v] = D[M=v + 8][N=lane-16]
    // diag(m): m in 0..7  -> lane m,    c[m]
    //          m in 8..15 -> lane m+16, c[m-8]
    const int e = lane & 7;
    float dot = c[0];
    dot = (e == 1) ? c[1] : dot;
    dot = (e == 2) ? c[2] : dot;
    dot = (e == 3) ? c[3] : dot;
    dot = (e == 4) ? c[4] : dot;
    dot = (e == 5) ? c[5] : dot;
    dot = (e == 6) ? c[6] : dot;
    dot = (e == 7) ? c[7] : dot;

    const bool active = (lane < 8) || (lane >= 24);
    if (active) {
        const int m = (lane < 8) ? lane : (lane - 16);   // row this lane owns
        // For these lanes, row == m, so uidx/iidx already match row m.
        float p = user_bias[uidx] + item_bias[iidx] + bias[0] + dot;
        p = fminf(fmaxf(p, 1.0f), 5.0f);
        const int o = base + m;
        if (o < batch) out[o] = p;
    }
}

extern "C" void kernel_launch(void* const* d_in, const int* in_sizes, int n_in,
                              void* d_out, int out_size, void* d_ws, size_t ws_size,
                              hipStream_t stream) {
    const int*   users     = (const int*)  d_in[0];
    const int*   items     = (const int*)  d_in[1];
    const float* user_emb  = (const float*)d_in[2];
    const float* item_emb  = (const float*)d_in[3];
    const float* user_bias = (const float*)d_in[4];
    const float* item_bias = (const float*)d_in[5];
    const float* bias      = (const float*)d_in[6];
    float*       out       = (float*)d_out;

    const int batch = in_sizes[0];
    const int rows_per_block = 128;                  // 8 waves * 16 rows
    const int grid = (batch + rows_per_block - 1) / rows_per_block;

    mf_wmma_dot_kernel<<<grid, 256, 0, stream>>>(
        users, items, user_emb, item_emb, user_bias, item_bias, bias, out, batch);
}